// GraphAttentionLayerV3_18949395710595
// MI455X (gfx1250) — compile-verified
//
#include <hip/hip_runtime.h>
#include <hip/hip_bf16.h>

typedef float v2f __attribute__((ext_vector_type(2)));
typedef float v8f __attribute__((ext_vector_type(8)));

#define Bc   32
#define Nn   512
#define INF  128
#define OUTF 256
#define Hh   8
#define Ff   32
#define ALPHA 0.2f
#define ASTRIDE (Nn + 4)   // LDS pad: row stride 516 floats -> 4-bank skew, conflict-free

// leaky_relu(x, 0.2) == max(x, 0.2*x) for alpha in [0,1): 2 VALU ops
__device__ __forceinline__ float lrelu(float x) { return fmaxf(x, ALPHA * x); }

// CDNA5 async global -> LDS copy (16 bytes), tracked with ASYNCcnt.
// lds_off = byte offset within the block's LDS allocation (low 32 bits of the
// generic pointer: LDS aperture maps addr[31:0] -> LDS offset).
__device__ __forceinline__ void async_copy_b128(uint32_t lds_off, const float* gsrc) {
    asm volatile("global_load_async_to_lds_b128 %0, %1, off"
                 :: "v"(lds_off), "v"(gsrc)
                 : "memory");
}
__device__ __forceinline__ uint32_t lds_off32(const float* p) {
    return (uint32_t)(size_t)p;
}

// Issue async copies for adj rows j0..j0+15 into a padded LDS stripe buffer.
// 8192 floats / 128 threads / 4 per copy = 16 async ops per thread.
__device__ __forceinline__ void issue_adj_stripe(const float* __restrict__ adj,
                                                 int j0, float* stripe, int tid) {
    for (int idx = tid * 4; idx < 16 * Nn; idx += 512) {
        int r = idx >> 9, c = idx & 511;
        async_copy_b128(lds_off32(&stripe[r * ASTRIDE + c]), &adj[(j0 + r) * Nn + c]);
    }
}

// ---------------------------------------------------------------------------
// Kernel 1: g = h @ W.  rows = B*N = 16384, K = 128, cols = 256.
// One wave per 16x16 output tile, f32 WMMA 16x16x4, K-loop of 32 steps.
// ---------------------------------------------------------------------------
__global__ __launch_bounds__(128) void k_gemm_g(const float* __restrict__ h,
                                                const float* __restrict__ W,
                                                float* __restrict__ g) {
    const int wave = threadIdx.x >> 5;
    const int lane = threadIdx.x & 31;
    const int tile = blockIdx.x * 4 + wave;           // 0 .. 16383
    const int r0 = (tile >> 4) * 16;                  // row tile (16384 rows)
    const int c0 = (tile & 15) * 16;                  // col tile (256 cols)
    const int m16   = lane & 15;
    const int khalf = (lane >> 4) * 2;                // 0 or 2

    v8f acc = {};
    for (int k0 = 0; k0 < INF; k0 += 4) {
        v2f a, b;
        a.x = h[(r0 + m16) * INF + k0 + khalf];
        a.y = h[(r0 + m16) * INF + k0 + khalf + 1];
        b.x = W[(k0 + khalf) * OUTF + c0 + m16];
        b.y = W[(k0 + khalf + 1) * OUTF + c0 + m16];
        acc = __builtin_amdgcn_wmma_f32_16x16x4_f32(false, a, false, b,
                                                    (short)0, acc, false, false);
    }
    const int rr = r0 + (lane >> 4) * 8;
#pragma unroll
    for (int r = 0; r < 8; ++r)
        g[(rr + r) * OUTF + c0 + m16] = acc[r];
}

// ---------------------------------------------------------------------------
// Kernel 2: s_i[b,h,n] = g[b,n,h,:] . w[:F]; s_j likewise with w[F:].
// Stored [b,h,n] so later kernels stream contiguously in n.
// ---------------------------------------------------------------------------
__global__ void k_scores(const float* __restrict__ g,
                         const float* __restrict__ w_attn,
                         float* __restrict__ si, float* __restrict__ sj) {
    int idx = blockIdx.x * blockDim.x + threadIdx.x;      // over B*N*H
    if (idx >= Bc * Nn * Hh) return;
    int hh = idx % Hh;
    int n  = (idx / Hh) % Nn;
    int b  = idx / (Hh * Nn);
    const float* gp = g + ((b * Nn + n) * Hh + hh) * Ff;
    float a0 = 0.f, a1 = 0.f;
#pragma unroll
    for (int f = 0; f < Ff; ++f) {
        float gv = gp[f];
        a0 += gv * w_attn[f];
        a1 += gv * w_attn[Ff + f];
    }
    si[(b * Hh + hh) * Nn + n] = a0;
    sj[(b * Hh + hh) * Nn + n] = a1;
}

// ---------------------------------------------------------------------------
// Kernel 3: pass 1 — online column softmax stats over i.
// e2[i,j] = sum_k lrelu(si[i]+sj[k]) * adj[j,k]   (A generated on the fly,
// B = adj^T stripe async-staged into LDS once per block; j-tile fixed).
// Block = 4 waves; wave w handles i-tiles w, w+4, ... (8 tiles each).
// Outputs M[b,h,j] and 1/Z[b,h,j].
// ---------------------------------------------------------------------------
__global__ __launch_bounds__(128) void k_colstats(const float* __restrict__ adj,
                                                  const float* __restrict__ si,
                                                  const float* __restrict__ sj,
                                                  float* __restrict__ Mb,
                                                  float* __restrict__ Zb) {
    __shared__ __align__(16) float adjS[16 * ASTRIDE];
    __shared__ __align__(16) float sjS[Nn];
    __shared__ float redM[4][16];
    __shared__ float redZ[4][16];

    const int bh  = blockIdx.y;            // b*H + h
    const int j0  = blockIdx.x * 16;
    const int tid = threadIdx.x;
    const int wave = tid >> 5, lane = tid & 31;
    const int m16 = lane & 15, khalf = (lane >> 4) * 2;

    // async-stage adj rows j0..j0+15 (B-matrix, reused by all 32 i-tiles)
    issue_adj_stripe(adj, j0, adjS, tid);
    for (int idx = tid * 4; idx < Nn; idx += 512)
        *(float4*)&sjS[idx] = *(const float4*)&sj[bh * Nn + idx];
    asm volatile("s_wait_asynccnt 0x0" ::: "memory");
    __syncthreads();

    float runM = -__builtin_inff(), runZ = 0.f;
    for (int it = wave; it < Nn / 16; it += 4) {
        const int i0 = it * 16;
        const float siv = si[bh * Nn + i0 + m16];
        v8f e = {};
        for (int k0 = 0; k0 < Nn; k0 += 4) {
            v2f a, b;
            a.x = lrelu(siv + sjS[k0 + khalf]);
            a.y = lrelu(siv + sjS[k0 + khalf + 1]);
            b.x = adjS[m16 * ASTRIDE + k0 + khalf];
            b.y = adjS[m16 * ASTRIDE + k0 + khalf + 1];
            e = __builtin_amdgcn_wmma_f32_16x16x4_f32(false, a, false, b,
                                                      (short)0, e, false, false);
        }
        // column (over i) reduce: 8 VGPRs per lane, then lane-half swap (N same
        // for lanes L and L^16 in the C/D layout; rows differ by 8).
        float tm = e[0];
#pragma unroll
        for (int r = 1; r < 8; ++r) tm = fmaxf(tm, e[r]);
        tm = fmaxf(tm, __shfl_xor(tm, 16, 32));
        float nm = fmaxf(runM, tm);
        float ts = 0.f;
#pragma unroll
        for (int r = 0; r < 8; ++r) ts += __expf(e[r] - nm);
        ts += __shfl_xor(ts, 16, 32);
        runZ = runZ * __expf(runM - nm) + ts;
        runM = nm;
    }
    if (lane < 16) { redM[wave][lane] = runM; redZ[wave][lane] = runZ; }
    __syncthreads();
    if (tid < 16) {
        float m = redM[0][tid];
#pragma unroll
        for (int w = 1; w < 4; ++w) m = fmaxf(m, redM[w][tid]);
        float z = 0.f;
#pragma unroll
        for (int w = 0; w < 4; ++w) z += redZ[w][tid] * __expf(redM[w][tid] - m);
        Mb[bh * Nn + j0 + tid] = m;
        Zb[bh * Nn + j0 + tid] = 1.0f / z;     // store reciprocal
    }
}

// ---------------------------------------------------------------------------
// Kernel 4: pass 2 — recompute e2 tiles, apply p = exp(e2 - M[j]) / Z[j],
// then out[i,f] += p(16x16) @ g_tile(16x32) via two more WMMA chains.
// Block = 4 waves, each wave owns a distinct i-tile (block covers 64 rows of i)
// and accumulates over all 32 j-tiles.  The adj j-stripe is DOUBLE-BUFFERED
// with async global->LDS copies: while the 128-WMMA score chain runs on the
// current stripe, the next stripe streams in (ASYNCcnt in-order completion:
// s_wait_asynccnt 16 guarantees the current buffer's 16 copies are done).
// p tile is transposed D-layout -> A-layout through padded LDS.
// ---------------------------------------------------------------------------
__global__ __launch_bounds__(128) void k_attn_out(const float* __restrict__ adj,
                                                  const float* __restrict__ si,
                                                  const float* __restrict__ sj,
                                                  const float* __restrict__ g,
                                                  const float* __restrict__ Mb,
                                                  const float* __restrict__ Zb,
                                                  float* __restrict__ out) {
    __shared__ __align__(16) float adjS[2][16 * ASTRIDE];
    __shared__ __align__(16) float sjS[Nn];
    __shared__ float pS[4][16][17];

    const int bh = blockIdx.y;
    const int b = bh / Hh, hh = bh % Hh;
    const int tid = threadIdx.x, wave = tid >> 5, lane = tid & 31;
    const int i0 = blockIdx.x * 64 + wave * 16;
    const int m16 = lane & 15, khalf = (lane >> 4) * 2;

    for (int idx = tid * 4; idx < Nn; idx += 512)
        *(float4*)&sjS[idx] = *(const float4*)&sj[bh * Nn + idx];
    const float siv = si[bh * Nn + i0 + m16];

    // prologue: stage stripe for j-tile 0
    issue_adj_stripe(adj, 0, adjS[0], tid);

    v8f acc0 = {}, acc1 = {};                 // out[16i x f0..15], [16i x f16..31]
    for (int jt = 0; jt < Nn / 16; ++jt) {
        const int j0 = jt * 16;
        const float* cur = adjS[jt & 1];
        if (jt + 1 < Nn / 16) {
            // overwrites the buffer last read at jt-1; end-of-iter barrier
            // below guarantees all waves are done with it.
            issue_adj_stripe(adj, (jt + 1) * 16, adjS[(jt + 1) & 1], tid);
            asm volatile("s_wait_asynccnt 0x10" ::: "memory"); // cur stripe done
        } else {
            asm volatile("s_wait_asynccnt 0x0" ::: "memory");
        }
        __syncthreads();                      // cur stripe visible block-wide

        // --- score tile e2[i0.., j0..] over full k ---
        v8f e = {};
        for (int k0 = 0; k0 < Nn; k0 += 4) {
            v2f a, bf;
            a.x = lrelu(siv + sjS[k0 + khalf]);
            a.y = lrelu(siv + sjS[k0 + khalf + 1]);
            bf.x = cur[m16 * ASTRIDE + k0 + khalf];
            bf.y = cur[m16 * ASTRIDE + k0 + khalf + 1];
            e = __builtin_amdgcn_wmma_f32_16x16x4_f32(false, a, false, bf,
                                                      (short)0, e, false, false);
        }

        // --- softmax weight (column j = j0 + m16) and stage transposed tile ---
        const float Mv = Mb[bh * Nn + j0 + m16];
        const float iZ = Zb[bh * Nn + j0 + m16];
        const int rbase = (lane >> 4) * 8;
#pragma unroll
        for (int r = 0; r < 8; ++r)
            pS[wave][rbase + r][m16] = __expf(e[r] - Mv) * iZ;   // pS[i][j]

        // --- out += p @ G,  G[k][f] = g[b, j0+k, hh, f] ---
#pragma unroll
        for (int kk = 0; kk < 16; kk += 4) {
            v2f a2, b20, b21;
            a2.x = pS[wave][m16][kk + khalf];
            a2.y = pS[wave][m16][kk + khalf + 1];
            const float* gp0 = &g[((size_t)(b * Nn + j0 + kk + khalf) * Hh + hh) * Ff];
            const float* gp1 = &g[((size_t)(b * Nn + j0 + kk + khalf + 1) * Hh + hh) * Ff];
            b20.x = gp0[m16];      b20.y = gp1[m16];
            b21.x = gp0[16 + m16]; b21.y = gp1[16 + m16];
            acc0 = __builtin_amdgcn_wmma_f32_16x16x4_f32(false, a2, false, b20,
                                                         (short)0, acc0, false, false);
            acc1 = __builtin_amdgcn_wmma_f32_16x16x4_f32(false, a2, false, b21,
                                                         (short)0, acc1, false, false);
        }
        __syncthreads();   // all waves done reading cur before it is re-staged
    }

    // out[b, i, hh*F + f]
    const int rr = i0 + (lane >> 4) * 8;
#pragma unroll
    for (int r = 0; r < 8; ++r) {
        out[((size_t)(b * Nn + rr + r) * Hh + hh) * Ff + m16]      = acc0[r];
        out[((size_t)(b * Nn + rr + r) * Hh + hh) * Ff + 16 + m16] = acc1[r];
    }
}

// ---------------------------------------------------------------------------
extern "C" void kernel_launch(void* const* d_in, const int* in_sizes, int n_in,
                              void* d_out, int out_size, void* d_ws, size_t ws_size,
                              hipStream_t stream) {
    (void)in_sizes; (void)n_in; (void)out_size; (void)ws_size;
    const float* h      = (const float*)d_in[0];   // [32,512,128]
    const float* W      = (const float*)d_in[1];   // [128,256]
    const float* w_attn = (const float*)d_in[2];   // [64]
    const float* adj    = (const float*)d_in[3];   // [512,512]
    float* out = (float*)d_out;                    // [32,512,256]

    float* ws = (float*)d_ws;                      // ~19 MB total
    float* g  = ws;                                // B*N*OUTF = 4,194,304
    float* si = g  + (size_t)Bc * Nn * OUTF;       // B*H*N    =   131,072
    float* sj = si + (size_t)Bc * Hh * Nn;
    float* Mb = sj + (size_t)Bc * Hh * Nn;
    float* Zb = Mb + (size_t)Bc * Hh * Nn;

    // 1) g = h @ W    (16384 x 128 x 256)
    k_gemm_g<<<dim3((Bc * Nn / 16) * (OUTF / 16) / 4), dim3(128), 0, stream>>>(h, W, g);
    // 2) attention score vectors
    k_scores<<<dim3((Bc * Nn * Hh) / 256), dim3(256), 0, stream>>>(g, w_attn, si, sj);
    // 3) pass 1: column softmax stats (M, 1/Z) per (b,h,j)
    k_colstats<<<dim3(Nn / 16, Bc * Hh), dim3(128), 0, stream>>>(adj, si, sj, Mb, Zb);
    // 4) pass 2: recompute scores, softmax-weight, second GEMM -> out
    k_attn_out<<<dim3(Nn / 64, Bc * Hh), dim3(128), 0, stream>>>(adj, si, sj, g, Mb, Zb, out);
}